// LowRankKVCache_88562225643882
// MI455X (gfx1250) — compile-verified
//
#include <hip/hip_runtime.h>
#include <hip/hip_bf16.h>
#include <cstdint>
#include <math.h>

// ---------------------------------------------------------------------------
// Types for CDNA5 WMMA / TDM builtins
// ---------------------------------------------------------------------------
typedef __attribute__((ext_vector_type(2))) float        v2f;
typedef __attribute__((ext_vector_type(8))) float        v8f;
typedef __attribute__((ext_vector_type(4))) unsigned int u32x4;
typedef __attribute__((ext_vector_type(4))) int          i32x4;
typedef __attribute__((ext_vector_type(8))) int          i32x8;

#if defined(__has_include)
#if __has_include(<hip/amd_detail/amd_gfx1250_TDM.h>)
#define TDM_ARGS6 1
#else
#define TDM_ARGS6 0
#endif
#else
#define TDM_ARGS6 0
#endif

// ---------------------------------------------------------------------------
// TDM: DMA `nrows` consecutive rows of a row-major [*,128] fp32 tensor into LDS.
// Descriptor per CDNA5 ISA ch.10.8 (D# group0/group1).
// ---------------------------------------------------------------------------
__device__ __forceinline__ void tdm_load_rows_128(const float* gsrc,
                                                  const float* lds_dst,
                                                  int nrows) {
  unsigned long long ga = (unsigned long long)(uintptr_t)gsrc;
  unsigned lds_off = (unsigned)(uintptr_t)lds_dst;   // flat LDS addr low 32 = LDS byte offset
  u32x4 g0;
  g0[0] = 1u;                                         // count=1 (valid user descriptor)
  g0[1] = lds_off;                                    // lds_addr [63:32]
  g0[2] = (unsigned)ga;                               // global_addr [95:64]
  g0[3] = (unsigned)((ga >> 32) & 0x01FFFFFFull)      // global_addr [120:96]
        | (2u << 30);                                 // type=2 ("image") [127:126]
  i32x8 g1;
  g1[0] = (2 << 16);                 // workgroup_mask=0, data_size=2 (4 bytes)
  g1[1] = (128 << 16);               // tensor_dim0 = 128 elements (bits 79:48 lo)
  g1[2] = (nrows & 0xFFFF) << 16;    // tensor_dim1 lo16 (bits 111:80)
  g1[3] = (128 << 16);               // tile_dim0 = 128 (bits 127:112)
  g1[4] = (nrows & 0xFFFF);          // tile_dim1 = nrows (bits 143:128)
  g1[5] = 128;                       // tensor_dim0_stride = 128 elements (bits 207:160 lo)
  g1[6] = 0;
  g1[7] = 0;
  i32x4 z4 = {0, 0, 0, 0};
#if TDM_ARGS6
  i32x8 z8 = {0, 0, 0, 0, 0, 0, 0, 0};
  __builtin_amdgcn_tensor_load_to_lds(g0, g1, z4, z4, z8, 0);
#else
  __builtin_amdgcn_tensor_load_to_lds(g0, g1, z4, z4, 0);
#endif
}

// ---------------------------------------------------------------------------
// fp32 WMMA: D(16x16) = A(16x4) * B(4x16) + C
// ---------------------------------------------------------------------------
__device__ __forceinline__ v8f wmma_f32(v2f a, v2f b, v8f c) {
  return __builtin_amdgcn_wmma_f32_16x16x4_f32(false, a, false, b,
                                               (short)0, c, false, false);
}

// ---------------------------------------------------------------------------
// K0: zero the Gram workspace (atomics accumulate into it).
// ---------------------------------------------------------------------------
__global__ void k0_zero(float* __restrict__ g, int n) {
  int i = blockIdx.x * blockDim.x + threadIdx.x;
  if (i < n) g[i] = 0.0f;
}

// ---------------------------------------------------------------------------
// K1: split-K Gram. grid = (8 splits, 64 matrices), 256 threads (8 waves).
// Each WG accumulates G += X_chunkᵀ X_chunk over 256 rows, TDM-double-buffered,
// then atomically adds its 128x128 partial into the workspace.
// Dynamic LDS: 2 * 64*128 floats = 64 KB.
// ---------------------------------------------------------------------------
__global__ void __launch_bounds__(256)
k1_gram(const float* __restrict__ x, float* __restrict__ gram) {
  const int mat   = blockIdx.y;
  const int split = blockIdx.x;
  const int tid   = threadIdx.x;
  const int wave  = tid >> 5;        // tile-row ti = wave (0..7)
  const int lane  = tid & 31;
  const int m     = lane & 15;       // M (A) / N (B,C) index
  const int kh    = lane >> 4;       // half-wave selector for K / M+8

  extern __shared__ float smem[];
  float* buf0 = smem;
  float* buf1 = smem + 64 * 128;
  const float* xm = x + (size_t)mat * 2048 * 128 + (size_t)split * 256 * 128;

  v8f acc[8] = {};

  if (tid < 32) tdm_load_rows_128(xm, buf0, 64);     // prologue: chunk 0
  for (int c = 0; c < 4; ++c) {                      // 4 chunks of 64 rows
    if (tid < 32) {
      if (c + 1 < 4) {
        tdm_load_rows_128(xm + (size_t)(c + 1) * 64 * 128,
                          ((c + 1) & 1) ? buf1 : buf0, 64);
        __builtin_amdgcn_s_wait_tensorcnt(1);        // chunk c landed
      } else {
        __builtin_amdgcn_s_wait_tensorcnt(0);
      }
    }
    __syncthreads();
    const float* xb = (c & 1) ? buf1 : buf0;
    for (int k0 = 0; k0 < 64; k0 += 4) {
      v2f a;   // A[m][k] = X[k0+k][wave*16+m]  (= Xᵀ tile)
      a.x = xb[(k0 + 2 * kh + 0) * 128 + wave * 16 + m];
      a.y = xb[(k0 + 2 * kh + 1) * 128 + wave * 16 + m];
#pragma unroll
      for (int tj = 0; tj < 8; ++tj) {
        v2f b; // B[k][n] = X[k0+k][tj*16+n]
        b.x = xb[(k0 + 2 * kh + 0) * 128 + tj * 16 + m];
        b.y = xb[(k0 + 2 * kh + 1) * 128 + tj * 16 + m];
        acc[tj] = wmma_f32(a, b, acc[tj]);
      }
    }
    __syncthreads();
  }

  float* gm = gram + (size_t)mat * 128 * 128;
#pragma unroll
  for (int tj = 0; tj < 8; ++tj)
#pragma unroll
    for (int r = 0; r < 8; ++r)
      atomicAdd(&gm[(wave * 16 + r + 8 * kh) * 128 + tj * 16 + m], acc[tj][r]);
}

// ---------------------------------------------------------------------------
// K2: parallel cyclic Jacobi eigensolver on G (128x128, in LDS) with
// eigenvector accumulation, top-64 selection, then P = Vsel·Vselᵀ via WMMA,
// written back over G in the workspace. grid = 64, 256 threads.
// Dynamic LDS: 128 KB (G + V).
// ---------------------------------------------------------------------------
__global__ void __launch_bounds__(256)
k2_eigs(float* __restrict__ gram) {
  const int mat = blockIdx.x;
  const int tid = threadIdx.x;
  extern __shared__ float smem[];
  float* G = smem;            // 128*128
  float* V = smem + 16384;    // 128*128
  __shared__ float rc[64], rs[64];
  __shared__ int   sel[64];
  float* gm = gram + (size_t)mat * 16384;

  if (tid < 32) {
    tdm_load_rows_128(gm, G, 128);
    __builtin_amdgcn_s_wait_tensorcnt(0);
  }
  for (int i = tid; i < 16384; i += 256)
    V[i] = ((i >> 7) == (i & 127)) ? 1.0f : 0.0f;
  __syncthreads();

  const int pi  = tid >> 2;   // pair id 0..63
  const int sub = tid & 3;    // 4 threads per pair
  for (int sweep = 0; sweep < 10; ++sweep) {
    for (int round = 0; round < 127; ++round) {
      // round-robin tournament: 64 disjoint (p,q) pairs per round
      int a = (pi == 0) ? 0 : ((round + pi - 1) % 127) + 1;
      int b = ((round + 126 - pi) % 127) + 1;
      int p = a < b ? a : b;
      int q = a < b ? b : a;
      if (sub == 0) {
        float app = G[p * 128 + p], aqq = G[q * 128 + q], apq = G[p * 128 + q];
        float c = 1.0f, s = 0.0f;
        if (fabsf(apq) > 1e-20f) {
          float tau = (aqq - app) / (2.0f * apq);
          float t = (tau >= 0.0f ? 1.0f : -1.0f) /
                    (fabsf(tau) + sqrtf(1.0f + tau * tau));
          c = rsqrtf(1.0f + t * t);
          s = t * c;
        }
        rc[pi] = c; rs[pi] = s;
      }
      __syncthreads();
      float c = rc[pi], s = rs[pi];
      // G <- Jᵀ G (disjoint rows across pairs)
      for (int j = sub * 32; j < sub * 32 + 32; ++j) {
        float gp = G[p * 128 + j], gq = G[q * 128 + j];
        G[p * 128 + j] = c * gp - s * gq;
        G[q * 128 + j] = s * gp + c * gq;
      }
      __syncthreads();
      // G <- G J, V <- V J (disjoint columns across pairs)
      for (int i = sub * 32; i < sub * 32 + 32; ++i) {
        float gp = G[i * 128 + p], gq = G[i * 128 + q];
        G[i * 128 + p] = c * gp - s * gq;
        G[i * 128 + q] = s * gp + c * gq;
        float vp = V[i * 128 + p], vq = V[i * 128 + q];
        V[i * 128 + p] = c * vp - s * vq;
        V[i * 128 + q] = s * vp + c * vq;
      }
      __syncthreads();
    }
  }

  // top-64 eigenvalue selection via rank counting on diag(G)
  if (tid < 128) {
    float ev = G[tid * 128 + tid];
    int rank = 0;
    for (int j = 0; j < 128; ++j) {
      float e = G[j * 128 + j];
      if (e > ev || (e == ev && j < tid)) ++rank;
    }
    if (rank < 64) sel[rank] = tid;
  }
  __syncthreads();

  // compact Vsel (128x64) into the (now dead) G region
  float* Vc = G;
  for (int i = tid; i < 128 * 64; i += 256) {
    int row = i >> 6, k = i & 63;
    Vc[row * 64 + k] = V[row * 128 + sel[k]];
  }
  __syncthreads();

  // P = Vc · Vcᵀ  (128x128, K=64) via WMMA; store over G in workspace
  const int wave = tid >> 5, lane = tid & 31, m = lane & 15, kh = lane >> 4;
  v8f acc[8] = {};
  for (int k0 = 0; k0 < 64; k0 += 4) {
    v2f av;
    av.x = Vc[(wave * 16 + m) * 64 + k0 + 2 * kh + 0];
    av.y = Vc[(wave * 16 + m) * 64 + k0 + 2 * kh + 1];
#pragma unroll
    for (int tj = 0; tj < 8; ++tj) {
      v2f bv; // B[k][n] = Vcᵀ[k][tj*16+n] = Vc[tj*16+n][k]
      bv.x = Vc[(tj * 16 + m) * 64 + k0 + 2 * kh + 0];
      bv.y = Vc[(tj * 16 + m) * 64 + k0 + 2 * kh + 1];
      acc[tj] = wmma_f32(av, bv, acc[tj]);
    }
  }
#pragma unroll
  for (int tj = 0; tj < 8; ++tj)
#pragma unroll
    for (int r = 0; r < 8; ++r)
      gm[(wave * 16 + r + 8 * kh) * 128 + tj * 16 + m] = acc[tj][r];
}

// ---------------------------------------------------------------------------
// K3: out = X · P. grid = (16 row-chunks, 64 matrices), 256 threads.
// X chunk (128x128) and P (128x128) TDM'd into LDS (128 KB); 8 waves x 8 tiles.
// X re-read hits L2 (whole tensor = 64 MB < 192 MB L2).
// ---------------------------------------------------------------------------
__global__ void __launch_bounds__(256)
k3_project(const float* __restrict__ x, const float* __restrict__ pmat,
           float* __restrict__ out) {
  const int mat = blockIdx.y;
  const int rcx = blockIdx.x;       // 128-row chunk index (0..15)
  const int tid = threadIdx.x;
  const int wave = tid >> 5, lane = tid & 31, m = lane & 15, kh = lane >> 4;
  extern __shared__ float smem[];
  float* Xs = smem;                 // 128*128
  float* Ps = smem + 16384;         // 128*128
  const float* xm = x + (size_t)mat * 2048 * 128 + (size_t)rcx * 128 * 128;

  if (tid < 32) {
    tdm_load_rows_128(xm, Xs, 128);
    tdm_load_rows_128(pmat + (size_t)mat * 16384, Ps, 128);
    __builtin_amdgcn_s_wait_tensorcnt(0);
  }
  __syncthreads();

  v8f acc[8] = {};
  for (int k0 = 0; k0 < 128; k0 += 4) {
    v2f a;   // A[m][k] = X[wave*16+m][k0+k]
    a.x = Xs[(wave * 16 + m) * 128 + k0 + 2 * kh + 0];
    a.y = Xs[(wave * 16 + m) * 128 + k0 + 2 * kh + 1];
#pragma unroll
    for (int tj = 0; tj < 8; ++tj) {
      v2f b; // B[k][n] = P[k0+k][tj*16+n]
      b.x = Ps[(k0 + 2 * kh + 0) * 128 + tj * 16 + m];
      b.y = Ps[(k0 + 2 * kh + 1) * 128 + tj * 16 + m];
      acc[tj] = wmma_f32(a, b, acc[tj]);
    }
  }

  float* om = out + (size_t)mat * 2048 * 128 + (size_t)rcx * 128 * 128;
#pragma unroll
  for (int tj = 0; tj < 8; ++tj)
#pragma unroll
    for (int r = 0; r < 8; ++r)
      om[(wave * 16 + r + 8 * kh) * 128 + tj * 16 + m] = acc[tj][r];
}

// ---------------------------------------------------------------------------
extern "C" void kernel_launch(void* const* d_in, const int* in_sizes, int n_in,
                              void* d_out, int out_size, void* d_ws, size_t ws_size,
                              hipStream_t stream) {
  (void)in_sizes; (void)n_in; (void)out_size; (void)ws_size;
  const float* x   = (const float*)d_in[0];   // kv_cache: 4*16*2048*128 fp32
  float*       out = (float*)d_out;           // same shape, fp32
  float*       ws  = (float*)d_ws;            // Gram/P workspace: 64*128*128 fp32 = 4 MB

  const int NG = 64 * 128 * 128;
  k0_zero<<<dim3((NG + 255) / 256), dim3(256), 0, stream>>>(ws, NG);
  k1_gram<<<dim3(8, 64),  dim3(256),  64 * 1024, stream>>>(x, ws);
  k2_eigs<<<dim3(64),     dim3(256), 128 * 1024, stream>>>(ws);
  k3_project<<<dim3(16, 64), dim3(256), 128 * 1024, stream>>>(x, ws, out);
}